// GCNDiagEncoder_68204080660516
// MI455X (gfx1250) — compile-verified
//
#include <hip/hip_runtime.h>

typedef __attribute__((ext_vector_type(2))) float v2f;
typedef __attribute__((ext_vector_type(8))) float v8f;

#define N_NODES 10000
#define N_EDGES 640000
#define D       128

__global__ __launch_bounds__(256) void zero_kernel(float* __restrict__ p, int n) {
    int i = blockIdx.x * blockDim.x + threadIdx.x;
    if (i < n) p[i] = 0.0f;
}

__global__ __launch_bounds__(256) void tanh_kernel(float* __restrict__ p, int n) {
    int i = blockIdx.x * blockDim.x + threadIdx.x;
    if (i < n) p[i] = tanhf(p[i]);
}

// One wave32 handles 16 edges. For a 16-edge tile:
//   OUT[16x128] = diag(val[0..15]) x B[16x128],  B[e][f] = x[src[e]][f] * W[f]
// computed as 8 feature tiles (N=16) x 4 K-chunks of V_WMMA_F32_16X16X4_F32.
// Because A is diagonal, every D element equals val_e * x[src_e][f] exactly
// (all other MAC terms are exact zeros), so this is bit-accurate f32.
// Rows of D are then scattered with native f32 atomics (segment sum in L2).
__global__ __launch_bounds__(256) void spmm_wmma_kernel(
    const float* __restrict__ xin,   // [N_NODES, D] input features (already tanh'd for layer 1)
    const int*   __restrict__ src,   // [N_EDGES]
    const int*   __restrict__ dst,   // [N_EDGES]
    const float* __restrict__ vals,  // [N_EDGES]
    const float* __restrict__ Wrow,  // [D] diagonal weights for this layer
    float*       __restrict__ out)   // [N_NODES, D] accumulator (pre-zeroed)
{
    const int lane  = threadIdx.x & 31;
    const int wave  = threadIdx.x >> 5;
    const int group = blockIdx.x * 8 + wave;   // 16 edges per group, 8 waves/block
    const int ebase = group * 16;
    const int lo    = lane & 15;               // 0..15
    const int half  = lane >> 4;               // 0 or 1

    // Each lane owns metadata of edge `lo` within the tile.
    const int   sidx = src[ebase + lo];
    const int   didx = dst[ebase + lo];
    const float vA   = vals[ebase + lo];

    // Redistribute src row indices to match the B-operand VGPR layout:
    // B VGPR0 holds K-row (2*half), VGPR1 holds K-row (2*half+1), per K-chunk kc.
    int s0[4], s1[4];
#pragma unroll
    for (int kc = 0; kc < 4; ++kc) {
        s0[kc] = __shfl(sidx, kc * 4 + 2 * half,     32);
        s1[kc] = __shfl(sidx, kc * 4 + 2 * half + 1, 32);
    }
    // D/C tile layout: VGPR r holds row (r + 8*half); fetch the dst node ids.
    int drow[8];
#pragma unroll
    for (int r = 0; r < 8; ++r)
        drow[r] = __shfl(didx, r + 8 * half, 32);

#pragma unroll
    for (int nt = 0; nt < 8; ++nt) {
        const int   f  = nt * 16 + lo;     // this lane's feature column
        const float wf = Wrow[f];

        v8f c = {};                        // f32 16x16 accumulator tile
#pragma unroll
        for (int kc = 0; kc < 4; ++kc) {
            // A = diag(val) chunk [16 x 4]: lane (half,lo) holds A[lo][2h],A[lo][2h+1]
            v2f a, b;
            a.x = (lo == kc * 4 + 2 * half    ) ? vA : 0.0f;
            a.y = (lo == kc * 4 + 2 * half + 1) ? vA : 0.0f;
            // B chunk [4 x 16]: gathered feature rows, W folded in
            b.x = xin[s0[kc] * D + f] * wf;
            b.y = xin[s1[kc] * D + f] * wf;
            c = __builtin_amdgcn_wmma_f32_16x16x4_f32(
                    /*neg_a=*/false, a, /*neg_b=*/false, b,
                    /*c_mod=*/(short)0, c, /*reuse_a=*/false, /*reuse_b=*/false);
        }
        // Scatter the 16 result rows (segment sum) with native f32 atomics.
#pragma unroll
        for (int r = 0; r < 8; ++r) {
            unsafeAtomicAdd(&out[drow[r] * D + f], c[r]);
        }
    }
}

extern "C" void kernel_launch(void* const* d_in, const int* in_sizes, int n_in,
                              void* d_out, int out_size, void* d_ws, size_t ws_size,
                              hipStream_t stream) {
    const float* x    = (const float*)d_in[0];
    const int*   src  = (const int*)  d_in[1];
    const int*   dst  = (const int*)  d_in[2];
    const float* vals = (const float*)d_in[3];
    const float* W    = (const float*)d_in[4];  // [2, D]
    float* out = (float*)d_out;
    float* h   = (float*)d_ws;                  // [N_NODES, D] inter-layer buffer

    const int n        = N_NODES * D;            // 1.28M elements
    const int eltBlks  = (n + 255) / 256;        // 5000
    const int spmmBlks = N_EDGES / (16 * 8);     // 5000 blocks, 8 waves x 16 edges

    // Layer 0: h = spmm(x * W0)
    zero_kernel<<<eltBlks, 256, 0, stream>>>(h, n);
    spmm_wmma_kernel<<<spmmBlks, 256, 0, stream>>>(x, src, dst, vals, W, h);
    // tanh between layers
    tanh_kernel<<<eltBlks, 256, 0, stream>>>(h, n);
    // Layer 1: out = spmm(tanh(h) * W1)
    zero_kernel<<<eltBlks, 256, 0, stream>>>(out, n);
    spmm_wmma_kernel<<<spmmBlks, 256, 0, stream>>>(h, src, dst, vals, W + D, out);
}